// LossModule_80831284511319
// MI455X (gfx1250) — compile-verified
//
#include <hip/hip_runtime.h>
#include <float.h>

// ---------------- problem constants (from reference) ----------------
#define N_PTS   65536
#define K_NN    12

// ---------------- NN search tiling ----------------
#define NN_BLOCK 256          // threads per block (8 waves)
#define NQ       4            // queries per thread (raises VALU:LDS ratio 4x)
#define QPB      (NN_BLOCK * NQ)        // 1024 queries per block
#define QBLOCKS  (N_PTS / QPB)          // 64
#define YCHUNKS  8                      // split candidate space across blocks
#define YSPAN    (N_PTS / YCHUNKS)      // 8192 candidates per block
#define TILE     2048                   // candidate tile in LDS (24 KB)
#define TILES_PER_CHUNK (YSPAN / TILE)  // 4
#define CHUNKS16_PER_TILE (TILE * 12 / 16 / NN_BLOCK)  // 6 async b128 per thread

#define FIN_BLOCKS (N_PTS / 256)        // 256

typedef __attribute__((ext_vector_type(2))) float v2f;
typedef __attribute__((ext_vector_type(8))) float v8f;

// ---------------- WMMA wave reduction ----------------
// Sum 32 lane values with one V_WMMA_F32_16X16X4_F32 against a ones B-matrix.
// A layout (32-bit A, 16x4): lanes 0-15 hold A[m][0] in a.x (a.y = A[m][1] = 0),
// lanes 16-31 hold A[m][2]. D[m][n] = v_m + v_{m+16} for every column n.
// sum of D VGPRs in lane 0  = rows 0..7, in lane 16 = rows 8..15.
__device__ __forceinline__ float wave_reduce_wmma(float v) {
  v2f a; a.x = v;    a.y = 0.0f;
  v2f b; b.x = 1.0f; b.y = 1.0f;
  v8f c = {};
  c = __builtin_amdgcn_wmma_f32_16x16x4_f32(false, a, false, b, (short)0, c,
                                            false, false);
  float t = ((c[0] + c[1]) + (c[2] + c[3])) + ((c[4] + c[5]) + (c[6] + c[7]));
  return __shfl(t, 0, 32) + __shfl(t, 16, 32);
}

// ---------------- CDNA5 async global->LDS copy of one candidate tile ----------------
__device__ __forceinline__ void prefetch_tile(const float* pc2, int yBase,
                                              int tile, unsigned ldsbase, int tid) {
  unsigned long long src = (unsigned long long)(size_t)pc2 +
                           (unsigned long long)(unsigned)(yBase + tile * TILE) * 12ull;
#pragma unroll
  for (int it = 0; it < CHUNKS16_PER_TILE; ++it) {
    unsigned off16 = (unsigned)(tid + it * NN_BLOCK) * 16u;
    asm volatile("global_load_async_to_lds_b128 %0, %1, off"
                 :: "v"(ldsbase + off16),
                    "v"(src + (unsigned long long)off16)
                 : "memory");
  }
}

// ---------------- init packed NN table ----------------
__global__ void init_nn(unsigned long long* __restrict__ pk) {
  int i = blockIdx.x * blockDim.x + threadIdx.x;
  if (i < N_PTS) pk[i] = 0xFFFFFFFFFFFFFFFFull;
}

// ---------------- brute-force L1 NN: x = pc1 + est_flow vs pc2 ----------------
__global__ __launch_bounds__(NN_BLOCK)
void nn_kernel(const float* __restrict__ pc1, const float* __restrict__ est_flow,
               const float* __restrict__ pc2, unsigned long long* __restrict__ pk) {
  __shared__ float ybuf[2][TILE * 3];   // 2 x 24 KB double buffer
  const int tid = threadIdx.x;

  float xq[NQ][3];
  float dm[NQ];
  unsigned im[NQ];
#pragma unroll
  for (int q = 0; q < NQ; ++q) {
    int gi = blockIdx.x * QPB + q * NN_BLOCK + tid;
    xq[q][0] = pc1[3 * gi + 0] + est_flow[3 * gi + 0];
    xq[q][1] = pc1[3 * gi + 1] + est_flow[3 * gi + 1];
    xq[q][2] = pc1[3 * gi + 2] + est_flow[3 * gi + 2];
    dm[q] = FLT_MAX;
    im[q] = 0u;
  }

  const int yBase = blockIdx.y * YSPAN;
  const unsigned lds0 = (unsigned)(size_t)(const void*)&ybuf[0][0];
  const unsigned lds1 = (unsigned)(size_t)(const void*)&ybuf[1][0];

  prefetch_tile(pc2, yBase, 0, lds0, tid);

  int p = 0;
  for (int t = 0; t < TILES_PER_CHUNK; ++t) {
    asm volatile("s_wait_asynccnt 0" ::: "memory");  // our tile-t loads done
    __syncthreads();                                  // everyone's tile-t visible;
                                                      // buffer 1-p no longer read
    if (t + 1 < TILES_PER_CHUNK)
      prefetch_tile(pc2, yBase, t + 1, p ? lds0 : lds1, tid);  // overlaps compute

    const float* yb = &ybuf[p][0];
    const unsigned yIdx0 = (unsigned)(yBase + t * TILE);
#pragma unroll 2
    for (int j = 0; j < TILE; ++j) {
      float y0 = yb[3 * j + 0];
      float y1 = yb[3 * j + 1];
      float y2 = yb[3 * j + 2];
      unsigned yIdx = yIdx0 + (unsigned)j;
#pragma unroll
      for (int q = 0; q < NQ; ++q) {
        float d = fabsf(xq[q][0] - y0) + fabsf(xq[q][1] - y1) +
                  fabsf(xq[q][2] - y2);
        if (d < dm[q]) { dm[q] = d; im[q] = yIdx; }   // strict '<' keeps first idx
      }
    }
    p ^= 1;
  }

  // combine across y-chunks: packed (dist|idx) min — monotone for dist >= 0,
  // tie-breaks to smallest index (matches argmin semantics). Deterministic.
#pragma unroll
  for (int q = 0; q < NQ; ++q) {
    int gi = blockIdx.x * QPB + q * NN_BLOCK + tid;
    unsigned long long v =
        ((unsigned long long)__float_as_uint(dm[q]) << 32) |
        (unsigned long long)im[q];
    atomicMin(&pk[gi], v);
  }
}

// ---------------- chamfer term + smoothness term, per point ----------------
__global__ __launch_bounds__(256)
void finalize_kernel(const float* __restrict__ pc1, const float* __restrict__ est_flow,
                     const float* __restrict__ pc2, const int* __restrict__ nn_idx,
                     const unsigned long long* __restrict__ pk,
                     float* __restrict__ partials) {
  int i = blockIdx.x * 256 + threadIdx.x;

  float f0 = est_flow[3 * i + 0];
  float f1 = est_flow[3 * i + 1];
  float f2 = est_flow[3 * i + 2];
  float x0 = pc1[3 * i + 0] + f0;
  float x1 = pc1[3 * i + 1] + f1;
  float x2 = pc1[3 * i + 2] + f2;

  unsigned idx = (unsigned)(pk[i] & 0xFFFFFFFFull);
  float cham = fabsf(x0 - pc2[3 * idx + 0]) + fabsf(x1 - pc2[3 * idx + 1]) +
               fabsf(x2 - pc2[3 * idx + 2]);

  float s = 0.0f;
#pragma unroll 1
  for (int k = 1; k < K_NN; ++k) {       // drop self column 0
    int j = nn_idx[i * K_NN + k];
    s += fabsf(f0 - est_flow[3 * j + 0]) + fabsf(f1 - est_flow[3 * j + 1]) +
         fabsf(f2 - est_flow[3 * j + 2]);
  }

  float local = cham * (1.0f / (float)N_PTS) +
                s * (1.0f / ((float)N_PTS * 11.0f));

  float ws = wave_reduce_wmma(local);     // EXEC all ones: exact launch, no divergence
  __shared__ float red[8];
  if ((threadIdx.x & 31) == 0) red[threadIdx.x >> 5] = ws;
  __syncthreads();
  if (threadIdx.x == 0) {
    float b = 0.0f;
    for (int w = 0; w < 8; ++w) b += red[w];
    partials[blockIdx.x] = b;             // deterministic block partial
  }
}

// ---------------- deterministic final sum of 256 block partials ----------------
__global__ __launch_bounds__(256)
void sum_kernel(const float* __restrict__ partials, float* __restrict__ out) {
  float v = partials[threadIdx.x];
  float ws = wave_reduce_wmma(v);
  __shared__ float red[8];
  if ((threadIdx.x & 31) == 0) red[threadIdx.x >> 5] = ws;
  __syncthreads();
  if (threadIdx.x == 0) {
    float b = 0.0f;
    for (int w = 0; w < 8; ++w) b += red[w];
    out[0] = b;
  }
}

extern "C" void kernel_launch(void* const* d_in, const int* in_sizes, int n_in,
                              void* d_out, int out_size, void* d_ws, size_t ws_size,
                              hipStream_t stream) {
  const float* pc1      = (const float*)d_in[0];
  const float* est_flow = (const float*)d_in[1];
  const float* pc2      = (const float*)d_in[2];
  const int*   nn_idx   = (const int*)d_in[3];
  float* out = (float*)d_out;

  // workspace layout: [0, 512KB) packed (dist|idx) per query; then block partials
  unsigned long long* pk = (unsigned long long*)d_ws;
  float* partials = (float*)((char*)d_ws + (size_t)N_PTS * sizeof(unsigned long long));

  init_nn<<<N_PTS / 256, 256, 0, stream>>>(pk);
  nn_kernel<<<dim3(QBLOCKS, YCHUNKS), NN_BLOCK, 0, stream>>>(pc1, est_flow, pc2, pk);
  finalize_kernel<<<FIN_BLOCKS, 256, 0, stream>>>(pc1, est_flow, pc2, nn_idx, pk, partials);
  sum_kernel<<<1, 256, 0, stream>>>(partials, out);
}